// QGate_65481071398618
// MI455X (gfx1250) — compile-verified
//
#include <hip/hip_runtime.h>

typedef __attribute__((ext_vector_type(2))) float v2f;
typedef __attribute__((ext_vector_type(8))) float v8f;

#define NQ    12
#define DEPTH 6
#define DIM   4096   // 2^12 amplitudes
#define NGRP  3      // 3 groups of 4 qubits

// ---------------------------------------------------------------------------
// Setup kernel: build 18 fused 16x16 complex group unitaries.
// For layer d, qubit q: U_q = RX(t2) * RZ(t1) * RY(t0)  (state <- RY,RZ,RX).
// Group matrix G = U_{4g} (x) U_{4g+1} (x) U_{4g+2} (x) U_{4g+3}, wire 4g = MSB.
// Output: uRe/uIm row-major [d*3+g][16][16] in d_ws.
// ---------------------------------------------------------------------------
__global__ void build_group_unitaries(const float* __restrict__ params,
                                      float* __restrict__ uRe,
                                      float* __restrict__ uIm) {
  const int tid = threadIdx.x;        // 256 threads: one (a,b) entry each
  const int a = tid >> 4, b = tid & 15;
  for (int d = 0; d < DEPTH; ++d) {
    for (int g = 0; g < NGRP; ++g) {
      float gr = 1.f, gi = 0.f;
      for (int j = 0; j < 4; ++j) {
        const int q = 4 * g + j;
        const float t0 = params[(d * NQ + q) * 3 + 0];
        const float t1 = params[(d * NQ + q) * 3 + 1];
        const float t2 = params[(d * NQ + q) * 3 + 2];
        const float cy = cosf(0.5f * t0), sy = sinf(0.5f * t0);
        const float cz = cosf(0.5f * t1), sz = sinf(0.5f * t1);
        const float cx = cosf(0.5f * t2), sx = sinf(0.5f * t2);
        // M1 = RZ * RY  (em = cz - i sz, ep = cz + i sz)
        const float m00r =  cy * cz, m00i = -cy * sz;
        const float m01r = -sy * cz, m01i =  sy * sz;
        const float m10r =  sy * cz, m10i =  sy * sz;
        const float m11r =  cy * cz, m11i =  cy * sz;
        // U = RX * M1, RX = [[cx, -i sx], [-i sx, cx]]
        const float u00r = cx * m00r + sx * m10i, u00i = cx * m00i - sx * m10r;
        const float u01r = cx * m01r + sx * m11i, u01i = cx * m01i - sx * m11r;
        const float u10r = sx * m00i + cx * m10r, u10i = -sx * m00r + cx * m10i;
        const float u11r = sx * m01i + cx * m11r, u11i = -sx * m01r + cx * m11i;
        const int aj = (a >> (3 - j)) & 1, bj = (b >> (3 - j)) & 1;
        float er, ei;
        if (aj == 0 && bj == 0)      { er = u00r; ei = u00i; }
        else if (aj == 0)            { er = u01r; ei = u01i; }
        else if (bj == 0)            { er = u10r; ei = u10i; }
        else                         { er = u11r; ei = u11i; }
        const float nr = gr * er - gi * ei;
        const float ni = gr * ei + gi * er;
        gr = nr; gi = ni;
      }
      const int base = (d * NGRP + g) * 256 + a * 16 + b;
      uRe[base] = gr;
      uIm[base] = gi;
    }
  }
}

// ---------------------------------------------------------------------------
// Main kernel: one block (8 wave32) per batch element, statevector in LDS.
// ---------------------------------------------------------------------------
__launch_bounds__(256)
__global__ void qsim_kernel(const float* __restrict__ x,
                            const float* __restrict__ uRe,
                            const float* __restrict__ uIm,
                            float* __restrict__ out) {
  extern __shared__ float lds[];          // 4 * DIM floats = 64 KB
  float* const bufRe0 = lds;
  float* const bufIm0 = lds + DIM;
  float* const bufRe1 = lds + 2 * DIM;
  float* const bufIm1 = lds + 3 * DIM;

  const int tid   = threadIdx.x;
  const int lane  = tid & 31;
  const int wave  = tid >> 5;
  const int batch = blockIdx.x;

  // --- init |0...0> ---
  #pragma unroll
  for (int k = 0; k < 16; ++k) {
    const int i = tid + k * 256;
    bufRe0[i] = 0.f; bufIm0[i] = 0.f;
  }
  __syncthreads();
  if (tid == 0) bufRe0[0] = 1.f;
  __syncthreads();

  // --- batched RX encoding (angle depends on batch element) ---
  for (int q = 0; q < NQ; ++q) {
    const float t = x[batch * NQ + q];
    const float c = cosf(0.5f * t), s = sinf(0.5f * t);
    const int sh = 11 - q;
    const int m = (1 << sh) - 1;
    #pragma unroll
    for (int k = 0; k < 8; ++k) {
      const int p  = tid + k * 256;                 // 2048 pairs
      const int i0 = ((p & ~m) << 1) | (p & m);
      const int i1 = i0 | (1 << sh);
      const float r0 = bufRe0[i0], q0 = bufIm0[i0];
      const float r1 = bufRe0[i1], q1 = bufIm0[i1];
      bufRe0[i0] = c * r0 + s * q1;
      bufIm0[i0] = c * q0 - s * r1;
      bufRe0[i1] = c * r1 + s * q0;
      bufIm0[i1] = c * q1 - s * r0;
    }
    __syncthreads();
  }

  int cur = 0;
  const int hm = lane & 15;   // element within lane-half
  const int h  = lane >> 4;   // lane half

  for (int d = 0; d < DEPTH; ++d) {
    // --- 3 group gates: D(16x256) = G(16x16) x S(16x256), complex, via WMMA ---
    for (int g = 0; g < NGRP; ++g) {
      float* const cRe = cur ? bufRe1 : bufRe0;
      float* const cIm = cur ? bufIm1 : bufIm0;
      float* const tRe = cur ? bufRe0 : bufRe1;
      float* const tIm = cur ? bufIm0 : bufIm1;
      const int lgR   = 8 - 4 * g;           // log2 of right-block size R
      const int R     = 1 << lgR;
      const int ubase = (d * NGRP + g) * 256;

      #pragma unroll
      for (int tt = 0; tt < 2; ++tt) {
        const int tile  = wave + tt * 8;     // 16 column-tiles over 8 waves
        const int col   = tile * 16 + hm;    // B/D column for this lane
        const int cbase = ((col >> lgR) << (lgR + 4)) | (col & (R - 1));

        v8f accRe = {};
        v8f accIm = {};
        #pragma unroll
        for (int kk = 0; kk < 4; ++kk) {
          const int ka = 4 * kk + 2 * h;     // A/B K indices for this lane-half
          // A operand: 16x4 slice of G (rows m=hm, K = ka, ka+1)
          const v2f aRe  = *(const v2f*)(uRe + ubase + hm * 16 + ka);
          const v2f aIm  = *(const v2f*)(uIm + ubase + hm * 16 + ka);
          const v2f aImN = -aIm;
          // B operand: 4x16 slice of state (rows K = ka, ka+1 at this column)
          v2f bRe, bIm;
          bRe.x = cRe[cbase + (ka    ) * R];
          bRe.y = cRe[cbase + (ka + 1) * R];
          bIm.x = cIm[cbase + (ka    ) * R];
          bIm.y = cIm[cbase + (ka + 1) * R];
          // complex GEMM: Dre = Gre*Sre - Gim*Sim ; Dim = Gre*Sim + Gim*Sre
          accRe = __builtin_amdgcn_wmma_f32_16x16x4_f32(
              false, aRe,  false, bRe, (short)0, accRe, false, false);
          accRe = __builtin_amdgcn_wmma_f32_16x16x4_f32(
              false, aImN, false, bIm, (short)0, accRe, false, false);
          accIm = __builtin_amdgcn_wmma_f32_16x16x4_f32(
              false, aRe,  false, bIm, (short)0, accIm, false, false);
          accIm = __builtin_amdgcn_wmma_f32_16x16x4_f32(
              false, aIm,  false, bRe, (short)0, accIm, false, false);
        }
        // D write-back: VGPR r holds row (r + 8*h) at this column
        #pragma unroll
        for (int r = 0; r < 8; ++r) {
          const int mo = r + 8 * h;
          tRe[cbase + mo * R] = accRe[r];
          tIm[cbase + mo * R] = accIm[r];
        }
      }
      __syncthreads();
      cur ^= 1;
    }

    // --- full CNOT ladder = wire-bit prefix-XOR permutation (one pass) ---
    {
      float* const cRe = cur ? bufRe1 : bufRe0;
      float* const cIm = cur ? bufIm1 : bufIm0;
      float* const tRe = cur ? bufRe0 : bufRe1;
      float* const tIm = cur ? bufIm0 : bufIm1;
      #pragma unroll
      for (int k = 0; k < 16; ++k) {
        const int i = tid + k * 256;
        int y = i;
        y ^= y >> 1; y ^= y >> 2; y ^= y >> 4; y ^= y >> 8;
        tRe[y] = cRe[i];
        tIm[y] = cIm[i];
      }
      __syncthreads();
      cur ^= 1;
    }
  }

  // --- Pauli-Z expectation values per wire ---
  float acc[NQ];
  #pragma unroll
  for (int w = 0; w < NQ; ++w) acc[w] = 0.f;
  {
    float* const cRe = cur ? bufRe1 : bufRe0;
    float* const cIm = cur ? bufIm1 : bufIm0;
    #pragma unroll
    for (int k = 0; k < 16; ++k) {
      const int i = tid + k * 256;
      const float re = cRe[i], im = cIm[i];
      const float p = re * re + im * im;
      #pragma unroll
      for (int w = 0; w < NQ; ++w)
        acc[w] += ((i >> (11 - w)) & 1) ? -p : p;
    }
  }
  // wave32 butterfly reduction
  #pragma unroll
  for (int off = 16; off >= 1; off >>= 1)
    #pragma unroll
    for (int w = 0; w < NQ; ++w)
      acc[w] += __shfl_xor(acc[w], off, 32);
  // combine 8 waves through (now free) other LDS buffer
  float* const red = cur ? bufRe0 : bufRe1;
  if (tid < NQ) red[tid] = 0.f;
  __syncthreads();
  if (lane == 0) {
    #pragma unroll
    for (int w = 0; w < NQ; ++w)
      atomicAdd(&red[w], acc[w]);
  }
  __syncthreads();
  if (tid < NQ) out[batch * NQ + tid] = red[tid];
}

// ---------------------------------------------------------------------------
extern "C" void kernel_launch(void* const* d_in, const int* in_sizes, int n_in,
                              void* d_out, int out_size, void* d_ws, size_t ws_size,
                              hipStream_t stream) {
  const float* x      = (const float*)d_in[0];   // (4096, 12) f32
  const float* params = (const float*)d_in[1];   // (6, 12, 3) f32
  float* out = (float*)d_out;                    // (4096, 12) f32

  float* uRe = (float*)d_ws;                     // 18 * 256 floats
  float* uIm = uRe + DEPTH * NGRP * 256;         // 18 * 256 floats

  build_group_unitaries<<<1, 256, 0, stream>>>(params, uRe, uIm);

  const size_t shmem = 4u * DIM * sizeof(float); // 64 KB double-buffered state
  qsim_kernel<<<4096, 256, shmem, stream>>>(x, uRe, uIm, out);
}